// Recurrent_Network_42786464203508
// MI455X (gfx1250) — compile-verified
//
#include <hip/hip_runtime.h>

typedef float v2f __attribute__((ext_vector_type(2)));
typedef float v8f __attribute__((ext_vector_type(8)));

#define N_COLS 1024
#define WID    128
#define DIN    1024
#define FDIM   (N_COLS * WID + N_COLS)   // 132096
#define SPLITS 32
#define CHUNK  (FDIM / SPLITS)           // 4128 floats = 16.5 KB LDS
#define WCHUNK (CHUNK / 8)               // 516 (8 waves per block)

static_assert(CHUNK * SPLITS == FDIM, "F split must be exact");
static_assert(WCHUNK % 4 == 0, "wave chunk must be multiple of K=4");

// D = A(16x4 f32) * B(4x16 f32) + C, exact f32 math on the matrix pipe.
__device__ __forceinline__ v8f wmma_mv_acc(v2f a, v2f b, v8f c) {
  return __builtin_amdgcn_wmma_f32_16x16x4_f32(false, a, false, b, (short)0, c,
                                               false, false);
}

// ---------------------------------------------------------------------------
// Phase 1a: f1[n,:] = relu(W1[n] @ x + b1[n])   (N blocks, 8 waves x 16 rows)
// ---------------------------------------------------------------------------
__global__ void __launch_bounds__(256)
k_f1(const float* __restrict__ W1, const float* __restrict__ x,
     const float* __restrict__ b1, float* __restrict__ f1) {
  __shared__ float xs[DIN];
  for (int i = threadIdx.x; i < DIN; i += 256) xs[i] = x[i];
  __syncthreads();

  const int n    = blockIdx.x;
  const int lane = threadIdx.x & 31;
  const int wave = threadIdx.x >> 5;
  const int row0 = wave * 16;
  const int m    = row0 + (lane & 15);
  // A layout: lanes 0-15 hold (K=0,1) of row m, lanes 16-31 hold (K=2,3)
  const float* wrow = W1 + ((size_t)n * WID + m) * DIN + ((lane < 16) ? 0 : 2);
  const bool  l0  = (lane == 0), l16 = (lane == 16);
  const int   sel = l16 ? 1 : 0;                   // lane16 reads odd K
  const float msk = (l0 || l16) ? 1.0f : 0.0f;     // everyone else -> 0
  v8f c = {};
#pragma unroll 8
  for (int k0 = 0; k0 < DIN; k0 += 4) {
    v2f a = *(const v2f*)(wrow + k0);
    v2f b;                                         // vector into B column 0
    b.x = xs[k0 + sel] * msk;                      // B[K0/K1][0]
    b.y = xs[k0 + 2 + sel] * msk;                  // B[K2/K3][0]
    c = wmma_mv_acc(a, b, c);
  }
  // D column 0: lane 0 has rows row0+0..7 in c[0..7], lane 16 rows row0+8..15
  if (l0 || l16) {
    const int ob = n * WID + row0 + (l16 ? 8 : 0);
    const float4 bb0 = *(const float4*)(b1 + ob);
    const float4 bb1 = *(const float4*)(b1 + ob + 4);
    float4 o0, o1;
    o0.x = fmaxf(c[0] + bb0.x, 0.0f);
    o0.y = fmaxf(c[1] + bb0.y, 0.0f);
    o0.z = fmaxf(c[2] + bb0.z, 0.0f);
    o0.w = fmaxf(c[3] + bb0.w, 0.0f);
    o1.x = fmaxf(c[4] + bb1.x, 0.0f);
    o1.y = fmaxf(c[5] + bb1.y, 0.0f);
    o1.z = fmaxf(c[6] + bb1.z, 0.0f);
    o1.w = fmaxf(c[7] + bb1.w, 0.0f);
    *(float4*)(f1 + ob)     = o0;
    *(float4*)(f1 + ob + 4) = o1;
  }
}

// ---------------------------------------------------------------------------
// Phase 1b: feat[n*W+:W] = relu(W2[n] @ f1[n] + b2[n])
// ---------------------------------------------------------------------------
__global__ void __launch_bounds__(256)
k_f2(const float* __restrict__ W2, const float* __restrict__ f1,
     const float* __restrict__ b2, float* __restrict__ feat) {
  __shared__ float vs[WID];
  if (threadIdx.x < WID) vs[threadIdx.x] = f1[(size_t)blockIdx.x * WID + threadIdx.x];
  __syncthreads();

  const int n    = blockIdx.x;
  const int lane = threadIdx.x & 31;
  const int wave = threadIdx.x >> 5;
  const int row0 = wave * 16;
  const int m    = row0 + (lane & 15);
  const float* wrow = W2 + ((size_t)n * WID + m) * WID + ((lane < 16) ? 0 : 2);
  const bool  l0  = (lane == 0), l16 = (lane == 16);
  const int   sel = l16 ? 1 : 0;
  const float msk = (l0 || l16) ? 1.0f : 0.0f;
  v8f c = {};
#pragma unroll 8
  for (int k0 = 0; k0 < WID; k0 += 4) {
    v2f a = *(const v2f*)(wrow + k0);
    v2f b;
    b.x = vs[k0 + sel] * msk;
    b.y = vs[k0 + 2 + sel] * msk;
    c = wmma_mv_acc(a, b, c);
  }
  if (l0 || l16) {
    const int ob = n * WID + row0 + (l16 ? 8 : 0);
    const float4 bb0 = *(const float4*)(b2 + ob);
    const float4 bb1 = *(const float4*)(b2 + ob + 4);
    float4 o0, o1;
    o0.x = fmaxf(c[0] + bb0.x, 0.0f);
    o0.y = fmaxf(c[1] + bb0.y, 0.0f);
    o0.z = fmaxf(c[2] + bb0.z, 0.0f);
    o0.w = fmaxf(c[3] + bb0.w, 0.0f);
    o1.x = fmaxf(c[4] + bb1.x, 0.0f);
    o1.y = fmaxf(c[5] + bb1.y, 0.0f);
    o1.z = fmaxf(c[6] + bb1.z, 0.0f);
    o1.w = fmaxf(c[7] + bb1.w, 0.0f);
    *(float4*)(feat + ob)     = o0;
    *(float4*)(feat + ob + 4) = o1;
  }
}

// ---------------------------------------------------------------------------
// Prep: feat tail = hidden_state, zero the phase-2 accumulator
// ---------------------------------------------------------------------------
__global__ void k_prep(const float* __restrict__ hidden,
                       float* __restrict__ feat, float* __restrict__ iacc) {
  const int i = blockIdx.x * blockDim.x + threadIdx.x;
  if (i < N_COLS) {
    feat[N_COLS * WID + i] = hidden[i];
    iacc[i] = 0.0f;
  }
}

// ---------------------------------------------------------------------------
// Phase 2: iacc[n] = sum_f Wi[n,f]*mask[n,f]*feat[f]
// grid (N/16, SPLITS); each block: 16 rows x one F-chunk, 8 waves split chunk
// ---------------------------------------------------------------------------
__global__ void __launch_bounds__(256)
k_phase2(const float* __restrict__ Wi, const float* __restrict__ mask,
         const float* __restrict__ feat, float* __restrict__ iacc) {
  __shared__ float fs[CHUNK];
  __shared__ float part[16];
  const int base = blockIdx.y * CHUNK;
  for (int i = threadIdx.x; i < CHUNK; i += 256) fs[i] = feat[base + i];
  if (threadIdx.x < 16) part[threadIdx.x] = 0.0f;
  __syncthreads();

  const int n0   = blockIdx.x * 16;
  const int lane = threadIdx.x & 31;
  const int wave = threadIdx.x >> 5;
  const int n    = n0 + (lane & 15);
  const size_t roff = (size_t)n * FDIM + base + ((lane < 16) ? 0 : 2);
  const float* wrow = Wi + roff;
  const float* mrow = mask + roff;
  const int kbeg = wave * WCHUNK;          // local (LDS) k index
  const int kend = kbeg + WCHUNK;
  const bool  l0  = (lane == 0), l16 = (lane == 16);
  const int   sel = l16 ? 1 : 0;
  const float msk = (l0 || l16) ? 1.0f : 0.0f;
  v8f c = {};
#pragma unroll 8
  for (int k0 = kbeg; k0 < kend; k0 += 4) {
    v2f wv = *(const v2f*)(wrow + k0);
    v2f mv = *(const v2f*)(mrow + k0);
    v2f a;
    a.x = wv.x * mv.x;
    a.y = wv.y * mv.y;
    v2f b;
    b.x = fs[k0 + sel] * msk;
    b.y = fs[k0 + 2 + sel] * msk;
    c = wmma_mv_acc(a, b, c);
  }
  if (l0 || l16) {
    const int rbase = l16 ? 8 : 0;
#pragma unroll
    for (int r = 0; r < 8; ++r) atomicAdd(&part[rbase + r], c[r]);
  }
  __syncthreads();
  if (threadIdx.x < 16) atomicAdd(&iacc[n0 + threadIdx.x], part[threadIdx.x]);
}

// ---------------------------------------------------------------------------
// Final: h = hidden + tanh(iacc + bi); y = sum(pred*h); out = [y, h]
// ---------------------------------------------------------------------------
__global__ void __launch_bounds__(1024)
k_final(const float* __restrict__ iacc, const float* __restrict__ bi,
        const float* __restrict__ hidden, const float* __restrict__ pred,
        float* __restrict__ out) {
  __shared__ float red[32];
  const int n = threadIdx.x;            // 1024 threads == N
  const float h = hidden[n] + tanhf(iacc[n] + bi[n]);
  out[1 + n] = h;
  float v = pred[n] * h;
  for (int off = 16; off > 0; off >>= 1) v += __shfl_down(v, off, 32);
  const int lane = n & 31, wave = n >> 5;
  if (lane == 0) red[wave] = v;
  __syncthreads();
  if (wave == 0) {
    float s = red[lane];
    for (int off = 16; off > 0; off >>= 1) s += __shfl_down(s, off, 32);
    if (lane == 0) out[0] = s;
  }
}

// ---------------------------------------------------------------------------
extern "C" void kernel_launch(void* const* d_in, const int* in_sizes, int n_in,
                              void* d_out, int out_size, void* d_ws,
                              size_t ws_size, hipStream_t stream) {
  const float* x      = (const float*)d_in[0];
  const float* hidden = (const float*)d_in[1];
  const float* W1     = (const float*)d_in[2];
  const float* b1     = (const float*)d_in[3];
  const float* W2     = (const float*)d_in[4];
  const float* b2     = (const float*)d_in[5];
  const float* Wi     = (const float*)d_in[6];
  const float* bi     = (const float*)d_in[7];
  const float* mask   = (const float*)d_in[8];
  const float* pred   = (const float*)d_in[9];
  float* out = (float*)d_out;

  float* ws   = (float*)d_ws;
  float* f1   = ws;                    // N*W      = 131072 floats
  float* feat = ws + N_COLS * WID;     // F        = 132096 floats
  float* iacc = feat + FDIM;           // N        = 1024 floats

  k_prep<<<4, 256, 0, stream>>>(hidden, feat, iacc);
  k_f1<<<N_COLS, 256, 0, stream>>>(W1, x, b1, f1);
  k_f2<<<N_COLS, 256, 0, stream>>>(W2, f1, b2, feat);
  k_phase2<<<dim3(N_COLS / 16, SPLITS), 256, 0, stream>>>(Wi, mask, feat, iacc);
  k_final<<<1, 1024, 0, stream>>>(iacc, bi, hidden, pred, out);
}